// HeteroGNNNoEmbedding_8418135900643
// MI455X (gfx1250) — compile-verified
//
#include <hip/hip_runtime.h>

// ---------------------------------------------------------------------------
// HeteroGNN (NNConv, max aggregation) for MI455X / gfx1250.
//
// Roofline: dominant cost = 4 passes of a [100000 x 1056] x [1056 x 32] f32
// GEMM (~26 GFLOP total); working set ~25 MB (L2-resident, HBM ~ microseconds
// at 23.3 TB/s) -> compute bound -> route through V_WMMA_F32_16X16X4_F32
// (full f32 precision, matches the f32 reference).
// Per-edge weight matrices are never materialized: msg = P @ B with
// P[e, h*32+d] = hid[e,h]*src_h[e,d] formed on the fly in registers, and the
// shared B (= mlp_w2 re-rowed + 32 bias rows, K=1056) staged once per block
// into LDS (135 KB, legal under CDNA5's 320 KB/WGP LDS).
// ---------------------------------------------------------------------------

#define EMB 32
#define OUTC 32
#define NEF 8
#define KTOT 1056              // EMB*OUTC matrix rows + EMB bias rows
#define KSTEPS (KTOT / 4)      // 264 K-steps of 4 for 16x16x4 WMMA
#define WAVES_PER_BLOCK 4
#define MSG_BLOCK (WAVES_PER_BLOCK * 32)

typedef __attribute__((ext_vector_type(2))) float v2f;
typedef __attribute__((ext_vector_type(8))) float v8f;

// Float atomic max via int atomics (valid for all sign combinations given the
// buffer is initialized to -inf; lowers to global_atomic_max_i32/min_u32).
__device__ __forceinline__ void atomic_max_f32(float* addr, float val) {
    if (val >= 0.0f)
        atomicMax((int*)addr, __float_as_int(val));
    else
        atomicMin((unsigned int*)addr, (unsigned int)__float_as_int(val));
}

// h[t,n,o] = sum_d x[t,n,d] * W[t,d,o] + b[t,o]
__global__ void node_lin_kernel(const float* __restrict__ x,
                                const float* __restrict__ w,
                                const float* __restrict__ b,
                                float* __restrict__ h, int N) {
    int gid = blockIdx.x * blockDim.x + threadIdx.x;
    int total = 2 * N * EMB;
    if (gid >= total) return;
    int o = gid & 31;
    int n = (gid >> 5) % N;
    int t = gid / (N * EMB);
    const float* xr = x + ((size_t)t * N + n) * EMB;
    const float* wr = w + (size_t)t * EMB * EMB;
    float acc = b[t * EMB + o];
#pragma unroll
    for (int d = 0; d < EMB; ++d) acc = fmaf(xr[d], wr[d * EMB + o], acc);
    h[gid] = acc;
}

// hid[e,o] = relu(edge_attr[e,:] @ w1 + b1)
__global__ void edge_mlp_kernel(const float* __restrict__ ea,
                                const float* __restrict__ w1,
                                const float* __restrict__ b1,
                                float* __restrict__ hid, int E) {
    int gid = blockIdx.x * blockDim.x + threadIdx.x;
    if (gid >= E * OUTC) return;
    int o = gid & 31;
    int e = gid >> 5;
    const float* er = ea + (size_t)e * NEF;
    float acc = b1[o];
#pragma unroll
    for (int j = 0; j < NEF; ++j) acc = fmaf(er[j], w1[j * OUTC + o], acc);
    hid[gid] = fmaxf(acc, 0.0f);
}

__global__ void fill_kernel(float* __restrict__ p, float v, int n) {
    int gid = blockIdx.x * blockDim.x + threadIdx.x;
    if (gid < n) p[gid] = v;
}

// The hot kernel: per 16-edge tile, msg[16,32] = P[16,1056] @ B[1056,32]
// via V_WMMA_F32_16X16X4_F32, scatter-max into agg.
__global__ __launch_bounds__(MSG_BLOCK, 1)
void msg_wmma_kernel(const float* __restrict__ hid,
                     const float* __restrict__ hsrc,
                     const int* __restrict__ esrc,
                     const int* __restrict__ edst,
                     const float* __restrict__ w2,
                     const float* __restrict__ b2,
                     float* __restrict__ agg, int E) {
    // B4[ks][n][j] = B[k=ks*4+j][n]; [n][4] blocking => conflict-free b64 loads.
    __shared__ float B4[KSTEPS][32][4];                 // 135168 B
    __shared__ float hid_s[WAVES_PER_BLOCK][16][33];    // padded rows
    __shared__ float src_s[WAVES_PER_BLOCK][16][34];    // padded rows
    __shared__ int   dst_s[WAVES_PER_BLOCK][16];

    // Stage B once per block: rows k<1024 from w2 (row r=h*32+d holds
    // w2[h, d*32+o]); rows 1024..1055 are the bias (src_h @ b2r term).
    for (int idx = threadIdx.x; idx < KTOT * 32; idx += MSG_BLOCK) {
        int k = idx >> 5, n = idx & 31;
        float v;
        if (k < EMB * OUTC)
            v = w2[(size_t)(k >> 5) * (EMB * OUTC) + (k & 31) * OUTC + n];
        else
            v = b2[(k - EMB * OUTC) * OUTC + n];
        B4[k >> 2][n][k & 3] = v;
    }
    __syncthreads();

    const int wave  = threadIdx.x >> 5;
    const int lane  = threadIdx.x & 31;
    const int m     = lane & 15;       // edge-in-tile / N-in-tile for this lane
    const int hi16  = lane >> 4;       // 0: K+{0,1}, 1: K+{2,3} half of wave
    const int koff  = hi16 * 2;
    const int ntiles = (E + 15) >> 4;
    const int gwave  = blockIdx.x * WAVES_PER_BLOCK + wave;
    const int stride = gridDim.x * WAVES_PER_BLOCK;

    for (int tile = gwave; tile < ntiles; tile += stride) {
        // Stage this wave's 16-edge tile (clamp tail: duplicate edges are
        // idempotent under max-aggregation).
        for (int i = 0; i < 16; ++i) {
            int e = tile * 16 + i;
            if (e > E - 1) e = E - 1;
            hid_s[wave][i][lane] = hid[(size_t)e * 32 + lane];
            int s = esrc[e];
            src_s[wave][i][lane] = hsrc[(size_t)s * 32 + lane];
        }
        if (lane < 16) {
            int e = tile * 16 + lane;
            if (e > E - 1) e = E - 1;
            dst_s[wave][lane] = edst[e];
        }
        // Same-wave LDS ops are in-order (DScnt) -> staging visible below.

        v8f c0 = {};  // output channels 0..15
        v8f c1 = {};  // output channels 16..31
        const float* srow = &src_s[wave][m][0];

#pragma unroll 1
        for (int h = 0; h < 33; ++h) {          // h==32 -> bias rows (hv=1)
            float hv = (h < 32) ? hid_s[wave][m][h] : 1.0f;
#pragma unroll
            for (int dk = 0; dk < 8; ++dk) {
                int ks = h * 8 + dk;
                int d0 = dk * 4 + koff;
                // A frag (16x4): lane l holds A[m][kbase+koff..+1]
                v2f a;
                a.x = hv * srow[d0];
                a.y = hv * srow[d0 + 1];
                // B frags (4x16): lane l holds B[kbase+koff..+1][n]
                v2f b0 = *(const v2f*)&B4[ks][m][koff];
                v2f b1 = *(const v2f*)&B4[ks][m + 16][koff];
                c0 = __builtin_amdgcn_wmma_f32_16x16x4_f32(
                        false, a, false, b0, (short)0, c0, false, false);
                c1 = __builtin_amdgcn_wmma_f32_16x16x4_f32(
                        false, a, false, b1, (short)0, c1, false, false);
            }
        }

        // C layout: vgpr i, lanes0-15 -> M=i, lanes16-31 -> M=i+8; N = lane&15.
#pragma unroll
        for (int i = 0; i < 8; ++i) {
            int mloc = hi16 * 8 + i;
            int nd = dst_s[wave][mloc];
            float* arow = agg + (size_t)nd * 32;
            atomic_max_f32(arow + m,      c0[i]);
            atomic_max_f32(arow + m + 16, c1[i]);
        }
    }
}

// out = relu( fix(agg) + h_dst @ root_w + conv_bias ),  fix(-inf)=0
__global__ void node_out_kernel(const float* __restrict__ agg,
                                const float* __restrict__ hdst,
                                const float* __restrict__ rw,
                                const float* __restrict__ cb,
                                float* __restrict__ out, int N) {
    int gid = blockIdx.x * blockDim.x + threadIdx.x;
    if (gid >= N * OUTC) return;
    int o = gid & 31;
    int n = gid >> 5;
    const float* hr = hdst + (size_t)n * EMB;
    float acc = cb[o];
#pragma unroll
    for (int d = 0; d < EMB; ++d) acc = fmaf(hr[d], rw[d * OUTC + o], acc);
    float a = agg[gid];
    if (a == -__builtin_inff()) a = 0.0f;  // empty segment -> 0 (PyG fill)
    out[gid] = fmaxf(a + acc, 0.0f);
}

extern "C" void kernel_launch(void* const* d_in, const int* in_sizes, int n_in,
                              void* d_out, int out_size, void* d_ws, size_t ws_size,
                              hipStream_t stream) {
    const float* x          = (const float*)d_in[0];
    const float* edge_attr  = (const float*)d_in[1];
    const int*   edge_src   = (const int*)  d_in[2];
    const int*   edge_dst   = (const int*)  d_in[3];
    const float* node_lin_w = (const float*)d_in[4];
    const float* node_lin_b = (const float*)d_in[5];
    const float* mlp_w1     = (const float*)d_in[6];
    const float* mlp_b1     = (const float*)d_in[7];
    const float* mlp_w2     = (const float*)d_in[8];
    const float* mlp_b2     = (const float*)d_in[9];
    const float* root_w     = (const float*)d_in[10];
    const float* conv_bias  = (const float*)d_in[11];
    (void)n_in; (void)out_size; (void)ws_size;

    const int N = in_sizes[0] / (2 * EMB);   // 20000
    const int E = in_sizes[2] / 2;           // 100000
    const int L = 2;

    float* ws  = (float*)d_ws;
    float* hA  = ws;                                   // [2,N,32]
    float* hB  = hA  + (size_t)2 * N * EMB;            // [2,N,32]
    float* hid = hB  + (size_t)2 * N * EMB;            // [E,32]
    float* agg = hid + (size_t)E * OUTC;               // [N,32]

    {
        int total = 2 * N * EMB;
        node_lin_kernel<<<(total + 255) / 256, 256, 0, stream>>>(
            x, node_lin_w, node_lin_b, hA, N);
    }

    for (int l = 0; l < L; ++l) {
        const float* hin = (l == 0) ? hA : hB;
        float* hout = (l == L - 1) ? (float*)d_out : hB;
        for (int t = 0; t < 2; ++t) {
            int dt = 1 - t;
            int lt = l * 2 + t;

            edge_mlp_kernel<<<(E * OUTC + 255) / 256, 256, 0, stream>>>(
                edge_attr + (size_t)t * E * NEF,
                mlp_w1 + (size_t)lt * NEF * OUTC,
                mlp_b1 + (size_t)lt * OUTC, hid, E);

            fill_kernel<<<(N * OUTC + 255) / 256, 256, 0, stream>>>(
                agg, -__builtin_inff(), N * OUTC);

            msg_wmma_kernel<<<200, MSG_BLOCK, 0, stream>>>(
                hid, hin + (size_t)t * N * EMB,
                edge_src + (size_t)t * E, edge_dst + (size_t)t * E,
                mlp_w2 + (size_t)lt * EMB * OUTC * OUTC,
                mlp_b2 + (size_t)lt * EMB * OUTC,
                agg, E);

            node_out_kernel<<<(N * OUTC + 255) / 256, 256, 0, stream>>>(
                agg, hin + (size_t)dt * N * EMB,
                root_w + (size_t)lt * EMB * OUTC,
                conv_bias + (size_t)lt * OUTC,
                hout + (size_t)dt * N * EMB, N);
        }
    }
}